// SimpleGNN_32066225832031
// MI455X (gfx1250) — compile-verified
//
#include <hip/hip_runtime.h>

typedef __attribute__((ext_vector_type(2))) float v2f;
typedef __attribute__((ext_vector_type(8))) float v8f;

#define D 128
#define WL_STRIDE 36   // 32 floats per K-panel row + pad (bank-conflict free, 16B-divisible)
#define A_STRIDE 260   // 256 floats (mean|x) + pad (bank-conflict free, 16B-divisible)

// ---- degree ---------------------------------------------------------------
__global__ void deg_kernel(const int* __restrict__ dst, float* __restrict__ deg, int E) {
  int e = blockIdx.x * blockDim.x + threadIdx.x;
  if (e < E) atomicAdd(&deg[dst[e]], 1.0f);
}

__global__ void invdeg_kernel(float* __restrict__ deg, int N) {
  int n = blockIdx.x * blockDim.x + threadIdx.x;
  if (n < N) deg[n] = 1.0f / fmaxf(deg[n], 1.0f);
}

// ---- scatter-add aggregation: wave per edge, float4 per lane --------------
__global__ __launch_bounds__(256) void scatter_kernel(
    const float* __restrict__ xin, const int* __restrict__ src,
    const int* __restrict__ dst, float* __restrict__ agg, int E)
{
  long long gid = (long long)blockIdx.x * blockDim.x + threadIdx.x;
  int e = (int)(gid >> 5);
  int lane = (int)(gid & 31);
  if (e >= E) return;
  int s = src[e], d = dst[e];
  float4 v = *(const float4*)(xin + (long long)s * D + lane * 4);
  float* p = agg + (long long)d * D + lane * 4;
  atomicAdd(p + 0, v.x);
  atomicAdd(p + 1, v.y);
  atomicAdd(p + 2, v.z);
  atomicAdd(p + 3, v.w);
}

// ---- fused SAGE layer: D = [mean | x] * [w_l^T ; w_r^T] + b, optional ReLU
// block = 256 threads = 8 waves; block owns 16 rows; wave w owns out cols [16w,16w+16)
__global__ __launch_bounds__(256) void sage_layer(
    const float* __restrict__ xin, const float* __restrict__ agg,
    const float* __restrict__ invdeg,
    const float* __restrict__ w_l, const float* __restrict__ w_r,
    const float* __restrict__ b_l, float* __restrict__ out, int N, int relu)
{
  __shared__ float lds_w[128 * WL_STRIDE];  // 18.4 KB: one 32-wide K panel of a weight matrix
  __shared__ float lds_a[16 * A_STRIDE];    // 16.6 KB: 16 rows of [mean(128) | x(128)]

  const int tid = threadIdx.x;
  const int row0 = blockIdx.x * 16;

  // Stage A rows (mean pre-scaled by invdeg), coalesced float4.
  for (int i = tid; i < 16 * 64; i += 256) {     // 64 float4 per 256-float row
    int r = i >> 6;
    int c4 = (i & 63) * 4;
    int row = row0 + r;
    float4 v = make_float4(0.f, 0.f, 0.f, 0.f);
    if (row < N) {
      if (c4 < 128) {
        v = *(const float4*)(agg + (long long)row * D + c4);
        float s = invdeg[row];
        v.x *= s; v.y *= s; v.z *= s; v.w *= s;
      } else {
        v = *(const float4*)(xin + (long long)row * D + (c4 - 128));
      }
    }
    *(float4*)&lds_a[r * A_STRIDE + c4] = v;
  }

  const int lane = tid & 31;
  const int wave = tid >> 5;
  const int col0 = wave * 16;
  const int m    = lane & 15;   // A: M row / B,D: N column within tile
  const int half = lane >> 4;   // K sub-pair select (K+0/1 vs K+2/3), D: M+8

  float bias = b_l[col0 + m];
  v8f c;
#pragma unroll
  for (int r = 0; r < 8; ++r) c[r] = bias;

  // K = 256 in 8 panels of 32; panels 0-3 from w_l, 4-7 from w_r.
  for (int p = 0; p < 8; ++p) {
    const float* wsrc = (p < 4) ? w_l : w_r;
    const int kbase = (p & 3) * 32;
    __syncthreads();  // previous panel's reads done (also orders A-staging for p==0)
    for (int i = tid; i < 1024; i += 256) {  // 128 rows x 8 float4
      int row = i >> 3;
      int c4 = (i & 7) * 4;
      *(float4*)&lds_w[row * WL_STRIDE + c4] =
          *(const float4*)(wsrc + row * D + kbase + c4);
    }
    __syncthreads();

    const float* arow = &lds_a[m * A_STRIDE] + p * 32 + 2 * half;
    const float* brow = &lds_w[(col0 + m) * WL_STRIDE] + 2 * half;
#pragma unroll
    for (int kl = 0; kl < 32; kl += 4) {
      v2f a = *(const v2f*)(arow + kl);  // A[m][k..k+1] (lanes 0-15) / [k+2..k+3]
      v2f b = *(const v2f*)(brow + kl);  // B[k..][col0+m] via w[o][k] contiguity
      c = __builtin_amdgcn_wmma_f32_16x16x4_f32(
          /*neg_a=*/false, a, /*neg_b=*/false, b,
          /*c_mod=*/(short)0, c, /*reuse_a=*/false, /*reuse_b=*/false);
    }
  }

  if (relu) {
#pragma unroll
    for (int r = 0; r < 8; ++r) c[r] = fmaxf(c[r], 0.0f);
  }

  const int rowb = row0 + 8 * half;  // D: lanes 16-31 hold M = r+8
#pragma unroll
  for (int r = 0; r < 8; ++r) {
    int row = rowb + r;
    if (row < N) out[(long long)row * D + col0 + m] = c[r];
  }
}

// ---- output head: h2 @ w_out^T + b_out (128 -> 4), thread per node --------
__global__ __launch_bounds__(256) void out_kernel(
    const float* __restrict__ h, const float* __restrict__ w_out,
    const float* __restrict__ b_out, float* __restrict__ out, int N)
{
  __shared__ float sw[4 * D];
  __shared__ float sb[4];
  for (int i = threadIdx.x; i < 4 * D; i += blockDim.x) sw[i] = w_out[i];
  if (threadIdx.x < 4) sb[threadIdx.x] = b_out[threadIdx.x];
  __syncthreads();

  int n = blockIdx.x * blockDim.x + threadIdx.x;
  if (n >= N) return;

  float a0 = sb[0], a1 = sb[1], a2 = sb[2], a3 = sb[3];
  const float4* row = (const float4*)(h + (long long)n * D);
#pragma unroll 4
  for (int q = 0; q < 32; ++q) {
    float4 v = row[q];
    int k = q * 4;
    a0 = fmaf(v.x, sw[0*D + k+0], a0); a0 = fmaf(v.y, sw[0*D + k+1], a0);
    a0 = fmaf(v.z, sw[0*D + k+2], a0); a0 = fmaf(v.w, sw[0*D + k+3], a0);
    a1 = fmaf(v.x, sw[1*D + k+0], a1); a1 = fmaf(v.y, sw[1*D + k+1], a1);
    a1 = fmaf(v.z, sw[1*D + k+2], a1); a1 = fmaf(v.w, sw[1*D + k+3], a1);
    a2 = fmaf(v.x, sw[2*D + k+0], a2); a2 = fmaf(v.y, sw[2*D + k+1], a2);
    a2 = fmaf(v.z, sw[2*D + k+2], a2); a2 = fmaf(v.w, sw[2*D + k+3], a2);
    a3 = fmaf(v.x, sw[3*D + k+0], a3); a3 = fmaf(v.y, sw[3*D + k+1], a3);
    a3 = fmaf(v.z, sw[3*D + k+2], a3); a3 = fmaf(v.w, sw[3*D + k+3], a3);
  }
  *(float4*)(out + (long long)n * 4) = make_float4(a0, a1, a2, a3);
}

// ---------------------------------------------------------------------------
extern "C" void kernel_launch(void* const* d_in, const int* in_sizes, int n_in,
                              void* d_out, int out_size, void* d_ws, size_t ws_size,
                              hipStream_t stream) {
  (void)n_in; (void)out_size; (void)ws_size;

  const float* x    = (const float*)d_in[0];
  const int*   ei   = (const int*)d_in[1];
  const float* w1_l = (const float*)d_in[2];
  const float* b1_l = (const float*)d_in[3];
  const float* w1_r = (const float*)d_in[4];
  const float* w2_l = (const float*)d_in[5];
  const float* b2_l = (const float*)d_in[6];
  const float* w2_r = (const float*)d_in[7];
  const float* w_o  = (const float*)d_in[8];
  const float* b_o  = (const float*)d_in[9];

  const int N = in_sizes[0] / D;   // 100000
  const int E = in_sizes[1] / 2;   // 1600000
  const int* src = ei;
  const int* dst = ei + E;

  float* agg  = (float*)d_ws;                    // N*128 f32
  float* h1   = agg  + (size_t)N * D;            // N*128 f32
  float* h2   = h1   + (size_t)N * D;            // N*128 f32
  float* ideg = h2   + (size_t)N * D;            // N f32 (deg -> invdeg in place)

  // degree / inverse degree (shared by both layers)
  hipMemsetAsync(ideg, 0, (size_t)N * sizeof(float), stream);
  deg_kernel<<<(E + 255) / 256, 256, 0, stream>>>(dst, ideg, E);
  invdeg_kernel<<<(N + 255) / 256, 256, 0, stream>>>(ideg, N);

  const int rowTiles = (N + 15) / 16;
  const long long sthreads = (long long)E * 32;
  const int sblocks = (int)((sthreads + 255) / 256);

  // layer 1
  hipMemsetAsync(agg, 0, (size_t)N * D * sizeof(float), stream);
  scatter_kernel<<<sblocks, 256, 0, stream>>>(x, src, dst, agg, E);
  sage_layer<<<rowTiles, 256, 0, stream>>>(x, agg, ideg, w1_l, w1_r, b1_l, h1, N, 1);

  // layer 2
  hipMemsetAsync(agg, 0, (size_t)N * D * sizeof(float), stream);
  scatter_kernel<<<sblocks, 256, 0, stream>>>(h1, src, dst, agg, E);
  sage_layer<<<rowTiles, 256, 0, stream>>>(h1, agg, ideg, w2_l, w2_r, b2_l, h2, N, 1);

  // output head
  out_kernel<<<(N + 255) / 256, 256, 0, stream>>>(h2, w_o, b_o, (float*)d_out, N);
}